// RPDC_3_19258633355554
// MI455X (gfx1250) — compile-verified
//
#include <hip/hip_runtime.h>

// RPDC depthwise 5x5 (16 sparse taps) over (16,256,128,128) fp32.
// Memory-bound (>=512MB compulsory HBM traffic -> ~22us @ 23.3TB/s, AI=4 FLOP/B).
// CDNA5 paths: TDM tensor_load_to_lds (one descriptor DMAs the whole 64KB slice
// HBM->LDS, hardware pad_enable creates the 136-float padded row stride) fenced
// by s_wait_tensorcnt; halo zeroing overlaps the DMA (disjoint LDS regions).

#define TILE_H   128
#define TILE_W   128
#define LDS_STRIDE 136            // floats per padded row; 136*4 % 16 == 0
#define LDS_ROWS   132            // 128 interior + 2 halo rows each side
#define LDS_FLOATS (LDS_ROWS * LDS_STRIDE)   // 17952 floats = 71808 B

typedef uint32_t u32x4 __attribute__((ext_vector_type(4)));
typedef uint32_t u32x8 __attribute__((ext_vector_type(8)));

__global__ __launch_bounds__(256) void rpdc_kernel(const float* __restrict__ x,
                                                   const float* __restrict__ wgt,
                                                   float* __restrict__ out,
                                                   int C)
{
    __shared__ __align__(16) float lds[LDS_FLOATS];

    const int tid   = threadIdx.x;
    const int slice = blockIdx.x;          // slice = n*C + c
    const int c     = slice % C;
    const float* xs = x   + (size_t)slice * (TILE_H * TILE_W);
    float*       os = out + (size_t)slice * (TILE_H * TILE_W);

    // ---- 1) one wave kicks off the TDM for the whole block ----
    // 2D tensor 128x128 fp32; tile == tensor; LDS dest padded to stride 136
    // floats via pad_interval=128 DWORDs (code 6), pad_amount=8 DWORDs (code 7).
    if (tid < 32) {
        const uint64_t ga = (uint64_t)(uintptr_t)xs;
        const uint32_t lds_dst =
            (uint32_t)(uintptr_t)(&lds[2 * LDS_STRIDE + 4]);   // interior origin
        u32x4 g0;
        g0.x = 1u;                                    // count=1: valid descriptor
        g0.y = lds_dst;                               // lds_addr (bytes)
        g0.z = (uint32_t)ga;                          // global_addr[31:0]
        g0.w = ((uint32_t)(ga >> 32) & 0x01FFFFFFu)   // global_addr[56:32]
             | 0x80000000u;                           // type=2 ("image")
        u32x8 g1;
        g1.s0 = (2u << 16)        // data_size = 4 bytes
              | (1u << 20)        // pad_enable
              | (6u << 22)        // pad_interval: 128 DWORDs
              | (7u << 25);       // pad_amount: 8 DWORDs -> row stride 136 floats
        g1.s1 = (128u << 16);     // tensor_dim0 = 128 (barrier addr = 0)
        g1.s2 = (128u << 16);     // tensor_dim1 = 128 (dim0 hi = 0)
        g1.s3 = (128u << 16);     // tile_dim0  = 128 (dim1 hi = 0)
        g1.s4 = 128u;             // tile_dim1 = 128, tile_dim2 = 0 (2D)
        g1.s5 = 128u;             // tensor_dim0_stride = 128
        g1.s6 = 0u;               // stride hi / dim1_stride lo (unused for 2D)
        g1.s7 = 0u;
        u32x4 gz = {0u, 0u, 0u, 0u};   // groups 2/3: tile_dim3/4=0 -> unused
        asm volatile("tensor_load_to_lds %0, %1, %2, %3"
                     :: "s"(g0), "s"(g1), "s"(gz), "s"(gz)
                     : "memory");
    }

    // ---- 2) zero ONLY the halo while the DMA streams (disjoint regions) ----
    {
        const float4 z = make_float4(0.f, 0.f, 0.f, 0.f);
        float4* p = (float4*)lds;                       // 34 float4 per row
        for (int i = tid; i < 136; i += 256) {          // rows 0,1,130,131 full
            const int r   = i / 34;
            const int q   = i % 34;
            const int row = (r < 2) ? r : 128 + r;      // 0,1,130,131
            p[row * 34 + q] = z;
        }
        {                                               // side quads, rows 2..129
            const int r    = 2 + (tid >> 1);            // 2..129
            const int side = tid & 1;
            p[r * 34 + (side ? 33 : 0)] = z;            // cols 0-3 / 132-135
        }
    }

    // ---- 3) per-channel taps w1..w8 (block-uniform -> scalarizable) ----
    float W[8];
    #pragma unroll
    for (int t = 0; t < 8; ++t) W[t] = wgt[c * 9 + 1 + t];

    if (tid < 32) {
#if defined(__has_builtin) && __has_builtin(__builtin_amdgcn_s_wait_tensorcnt)
        __builtin_amdgcn_s_wait_tensorcnt(0);
#else
        asm volatile("s_wait_tensorcnt 0" ::: "memory");
#endif
    }
    __syncthreads();   // publish TDM data + halo zeros to all waves

    // ---- 4) compute: 4-wide column strip, 16 rows, rolling 5x8 reg window ----
    const int w0 = (tid & 31) << 2;    // output col base 0..124
    const int h0 = (tid >> 5) << 4;    // output row base 0..112

    float win[5][8];
    // preload absolute rows h0-2 .. h0+1 (LDS rows h0 .. h0+3; halo rows are 0)
    #pragma unroll
    for (int i = 0; i < 4; ++i) {
        const int base = (h0 + i) * LDS_STRIDE + 2 + w0;   // col w0-2 at j=0
        #pragma unroll
        for (int j = 0; j < 8; ++j) win[i][j] = lds[base + j];
    }

    #pragma unroll
    for (int i = 0; i < 16; ++i) {
        // load absolute row h0+2+i into rotating slot
        {
            const int base = (h0 + 4 + i) * LDS_STRIDE + 2 + w0;
            float* dst = win[(4 + i) % 5];
            #pragma unroll
            for (int j = 0; j < 8; ++j) dst[j] = lds[base + j];
        }
        const int im2 = (0 + i) % 5;   // row h-2
        const int im1 = (1 + i) % 5;   // row h-1
        const int i0  = (2 + i) % 5;   // row h
        const int ip1 = (3 + i) % 5;   // row h+1
        const int ip2 = (4 + i) % 5;   // row h+2

        float acc[4];
        #pragma unroll
        for (int k = 0; k < 4; ++k) {
            const int m = k + 2;       // window col index of output pixel
            float a;
            a  = W[0] * (win[im2][m - 2] - win[im1][m - 1]);
            a += W[1] * (win[im2][m    ] - win[im1][m    ]);
            a += W[2] * (win[im2][m + 2] - win[im1][m + 1]);
            a += W[3] * (win[i0 ][m - 2] - win[i0 ][m - 1]);
            a += W[4] * (win[i0 ][m + 2] - win[i0 ][m + 1]);
            a += W[5] * (win[ip2][m - 2] - win[ip1][m - 1]);
            a += W[6] * (win[ip2][m    ] - win[ip1][m    ]);
            a += W[7] * (win[ip2][m + 2] - win[ip1][m + 1]);
            acc[k] = a;
        }
        *(float4*)(os + (h0 + i) * TILE_W + w0) =
            make_float4(acc[0], acc[1], acc[2], acc[3]);
    }
}

extern "C" void kernel_launch(void* const* d_in, const int* in_sizes, int n_in,
                              void* d_out, int out_size, void* d_ws, size_t ws_size,
                              hipStream_t stream) {
    const float* x = (const float*)d_in[0];
    const float* w = (const float*)d_in[1];
    float* out = (float*)d_out;
    const int slices = in_sizes[0] / (TILE_H * TILE_W);   // N*C = 4096
    const int C      = in_sizes[1] / 9;                   // 256
    rpdc_kernel<<<slices, 256, 0, stream>>>(x, w, out, C);
}